// Model_51582557225578
// MI455X (gfx1250) — compile-verified
//
#include <hip/hip_runtime.h>
#include <cmath>

// ---------------------------------------------------------------------------
// Spherical CNN (S2 conv -> SO3 conv -> SO3 conv -> mean) for MI455X / gfx1250.
//
// Roofline: spectral tensors are ~252 MB/layer; pipeline moves ~1.5-2 GB total
// -> ~80us at 23.3 TB/s HBM, while transform math is only ~30 GFLOP/layer.
// => HBM-bound. We keep full f32 precision (matches reference complex64) and
// use V_WMMA_F32_16X16X4_F32 for the batched 64x64 complex GEMMs implementing
// fft2/ifft2 (Y = W X W^T per plane), staged via LDS — filled with
// ASYNCcnt-tracked global_load_async_to_lds_b128 copies (CDNA5 async mover).
// Irregular per-l Wigner contractions stay on VALU.
// ---------------------------------------------------------------------------

#define PI_ 3.14159265358979323846
#define LSUM 43680           // sum_{l<32} (2l+1)^2
#define VOL  262144          // 64^3
#define BATCH 2
#define F0c 3
#define F1c 60
#define F2c 36

typedef float v2f __attribute__((ext_vector_type(2)));
typedef float v8f __attribute__((ext_vector_type(8)));

#if defined(__has_builtin)
#if __has_builtin(__builtin_amdgcn_global_load_async_to_lds_b128) && \
    __has_builtin(__builtin_amdgcn_s_wait_asynccnt)
#define HAS_ASYNC_LDS 1
#endif
#endif
#ifndef HAS_ASYNC_LDS
#define HAS_ASYNC_LDS 0
#endif

#if HAS_ASYNC_LDS
// builtin signature (from compiler diagnostic): param0 = int4 addrspace(1)*,
// param1 = LDS destination, then imm offset / imm cpol.
typedef int vi4 __attribute__((vector_size(4 * sizeof(int))));
typedef __attribute__((address_space(1))) vi4 g_vi4;
typedef __attribute__((address_space(3))) vi4 l_vi4;
__device__ __forceinline__ void async_cp16(const float* g, float* l) {
  __builtin_amdgcn_global_load_async_to_lds_b128((g_vi4*)g, (l_vi4*)l, 0, 0);
}
#endif

__device__ __forceinline__ v8f wmma4(v2f a, v2f b, v8f c) {
  // D = A(16x4,f32) * B(4x16,f32) + C(16x16,f32)
  return __builtin_amdgcn_wmma_f32_16x16x4_f32(false, a, false, b, (short)0, c,
                                               false, false);
}

__device__ __forceinline__ int l_off(int l) { return l * (4 * l * l - 1) / 3; }
__device__ __forceinline__ int decode_l(int idx) {
  int l = 0;
  while (l < 31 && l_off(l + 1) <= idx) ++l;
  return l;
}

// --------------------------- constant generation ---------------------------

// Wigner small-d for all l<32 at beta_j (j<64) and beta=pi/2 (block 64).
__global__ void k_wigner(float* dq, float* de) {
  __shared__ double lf[130];
  if (threadIdx.x == 0) {
    lf[0] = 0.0;
    for (int i = 1; i < 130; i++) lf[i] = lf[i - 1] + log((double)i);
  }
  __syncthreads();
  int j = blockIdx.x;
  double beta = (j < 64) ? (j + 0.5) * PI_ / 64.0 : PI_ * 0.5;
  float* out = (j < 64) ? dq + (size_t)j * LSUM : de;
  double cb = cos(0.5 * beta), sb = sin(0.5 * beta);
  double lc = log(cb), ls = log(sb);
  for (int idx = threadIdx.x; idx < LSUM; idx += blockDim.x) {
    int l = decode_l(idx);
    int rem = idx - l_off(l);
    int dim = 2 * l + 1;
    int mi = rem / dim, ni = rem % dim;
    int mm = mi - l, nn = ni - l;
    double pref = 0.5 * (lf[l + mm] + lf[l - mm] + lf[l + nn] + lf[l - nn]);
    int smin = (nn - mm > 0) ? nn - mm : 0;
    int smax = (l + nn < l - mm) ? l + nn : l - mm;
    double acc = 0.0;
    for (int s = smin; s <= smax; s++) {
      double lt = pref - lf[l + nn - s] - lf[s] - lf[mm - nn + s] -
                  lf[l - mm - s] + (double)(2 * l + nn - mm - 2 * s) * lc +
                  (double)(mm - nn + 2 * s) * ls;
      acc += (((mm - nn + s) & 1) ? -1.0 : 1.0) * exp(lt);
    }
    out[idx] = (float)acc;
  }
}

__global__ void k_dhw(float* wgt) {
  int j = threadIdx.x;
  if (j >= 64) return;
  double s = 0.0;
  for (int k = 0; k < 32; k++)
    s += sin((2.0 * j + 1.0) * (2.0 * k + 1.0) * PI_ / 128.0) / (2.0 * k + 1.0);
  wgt[j] = (float)(2.0 / 32.0 * sin(PI_ * (2.0 * j + 1.0) / 128.0) * s);
}

__global__ void k_twiddle(float* wfre, float* wfim, float* wire, float* wiim) {
  int idx = blockIdx.x * blockDim.x + threadIdx.x;
  if (idx >= 4096) return;
  int m = idx >> 6, a = idx & 63;
  double ang = -2.0 * PI_ * (double)(m * a) / 64.0;
  float c = (float)cos(ang), s = (float)sin(ang);
  wfre[idx] = c; wfim[idx] = s;   // forward DFT
  wire[idx] = c; wiim[idx] = -s;  // conj(F): ifft * N per axis
  // note: wiim == -wfim; we exploit this to avoid negation VALU in WMMA loops
}

// kernel spectrum: khat[i,o,mu] = scale * sum_p k[i,o,p] e^{-2pi i p mu/64}
__global__ void k_khat(const float* k, int Fi, int Fo, float scale,
                       const float* wfre, const float* wfim, float* khre,
                       float* khim) {
  int idx = blockIdx.x * blockDim.x + threadIdx.x;
  if (idx >= Fi * Fo * 64) return;
  int mu = idx & 63, row = idx >> 6;
  float ar = 0.f, ai = 0.f;
  for (int p = 0; p < 64; p++) {
    float v = k[row * 64 + p];
    ar += v * wfre[mu * 64 + p];
    ai += v * wfim[mu * 64 + p];
  }
  khre[idx] = ar * scale;
  khim[idx] = ai * scale;
}

// ------------------------------- layer 1 (S2) -------------------------------

__global__ void k_dft_alpha(const float* x, const float* wfre,
                            const float* wfim, float* xre, float* xim) {
  int idx = blockIdx.x * blockDim.x + threadIdx.x;
  if (idx >= BATCH * F0c * 64 * 64) return;
  int u = idx & 63, row = idx >> 6;
  const float* xr = x + row * 64;
  float ar = 0.f, ai = 0.f;
  for (int a = 0; a < 64; a++) {
    float v = xr[a];
    ar += v * wfre[u * 64 + a];
    ai += v * wfim[u * 64 + a];
  }
  xre[idx] = ar;
  xim[idx] = ai;
}

// fh[l,m,z,f] = sum_b (w[b]*dq[l][b,m,l]/(2N)) * xf[z,f,b,(m-l)%64]
__global__ void k_fh_s2(const float* xre, const float* xim, const float* dq,
                        const float* wgt, float* fh1) {
  int e = blockIdx.x * blockDim.x + threadIdx.x;
  if (e >= 1024 * BATCH * F0c) return;
  int zf = e % (BATCH * F0c);
  int lm = e / (BATCH * F0c);
  int l = 0;
  while (l < 31 && (l + 1) * (l + 1) <= lm) l++;
  int mi = lm - l * l, dim = 2 * l + 1;
  int u = (mi - l) & 63;
  const float C2 = 1.0f / 128.0f;
  float sr = 0.f, si = 0.f;
  for (int b = 0; b < 64; b++) {
    float wa = wgt[b] * dq[(size_t)b * LSUM + l_off(l) + mi * dim + l] * C2;
    int xi = (zf * 64 + b) * 64 + u;
    sr += wa * xre[xi];
    si += wa * xim[xi];
  }
  fh1[2 * e] = sr;
  fh1[2 * e + 1] = si;
}

// zs[l,m,n,z,o] = de[l][n,l] * sum_f fh[l,m,z,f] * conj(khat[f,o,(n-l)%64])
__global__ void k_zs_s2(const float* fh1, const float* de, const float* khre,
                        const float* khim, float* zs) {
  long long e = (long long)blockIdx.x * blockDim.x + threadIdx.x;
  const int C = BATCH * F1c;
  if (e >= (long long)LSUM * C) return;
  int zo = (int)(e % C);
  int lmn = (int)(e / C);
  int l = decode_l(lmn);
  int rem = lmn - l_off(l), dim = 2 * l + 1;
  int mi = rem / dim, ni = rem % dim;
  int z = zo / F1c, o = zo % F1c;
  int mu = (ni - l) & 63;
  float dn = de[l_off(l) + ni * dim + l];
  float sr = 0.f, si = 0.f;
  for (int f = 0; f < F0c; f++) {
    int fhidx = (l * l + mi) * (BATCH * F0c) + z * F0c + f;
    float ar = fh1[2 * fhidx], ai = fh1[2 * fhidx + 1];
    int kidx = (f * F1c + o) * 64 + mu;
    float cr = khre[kidx], ci = khim[kidx];
    sr += ar * cr + ai * ci;   // a * conj(c)
    si += ai * cr - ar * ci;
  }
  zs[2 * e] = dn * sr;
  zs[2 * e + 1] = dn * si;
}

// ------------------------------ SO3 layers ---------------------------------

// fh[l,m,k,z,f] = sum_b (w[b]*dq[l][b,m,k]/(2N^2)) * xf[z,f,b,(m-l)%64,(k-l)%64]
__global__ void k_fh_so3(const float* xre, const float* xim, const float* dq,
                         const float* wgt, int Fin, float* fh) {
  long long e = (long long)blockIdx.x * blockDim.x + threadIdx.x;
  const int C = BATCH * Fin;
  if (e >= (long long)LSUM * C) return;
  int zf = (int)(e % C);
  int lmk = (int)(e / C);
  int l = decode_l(lmk);
  int rem = lmk - l_off(l), dim = 2 * l + 1;
  int mi = rem / dim, ki = rem % dim;
  int um = (mi - l) & 63, uk = (ki - l) & 63;
  const float C3 = 1.0f / 8192.0f;
  float sr = 0.f, si = 0.f;
  for (int b = 0; b < 64; b++) {
    float wa = wgt[b] * dq[(size_t)b * LSUM + l_off(l) + mi * dim + ki] * C3;
    size_t xi = ((size_t)zf * 64 + b) * 4096 + (size_t)um * 64 + uk;
    sr += wa * xre[xi];
    si += wa * xim[xi];
  }
  fh[2 * e] = sr;
  fh[2 * e + 1] = si;
}

// g[l,m,n,z,f] = sum_k fh[l,m,k,z,f] * de[l][n,k]
__global__ void k_g_so3(const float* fh, const float* de, int Fin, float* g) {
  long long e = (long long)blockIdx.x * blockDim.x + threadIdx.x;
  const int C = BATCH * Fin;
  if (e >= (long long)LSUM * C) return;
  int zf = (int)(e % C);
  int lmn = (int)(e / C);
  int l = decode_l(lmn);
  int rem = lmn - l_off(l), dim = 2 * l + 1;
  int mi = rem / dim, ni = rem % dim;
  int base = l_off(l);
  float sr = 0.f, si = 0.f;
  for (int ki = 0; ki < dim; ki++) {
    float dnk = de[base + ni * dim + ki];
    long long fi = ((long long)(base + mi * dim + ki) * C + zf) * 2;
    sr += dnk * fh[fi];
    si += dnk * fh[fi + 1];
  }
  g[2 * e] = sr;
  g[2 * e + 1] = si;
}

// zs[l,m,n,z,o] = sum_f g[l,m,n,z,f] * conj(khat[f,o,(n-l)%64])
__global__ void k_zs_so3(const float* g, const float* khre, const float* khim,
                         int Fin, int Fo, float* zs) {
  long long e = (long long)blockIdx.x * blockDim.x + threadIdx.x;
  const int C = BATCH * Fo;
  if (e >= (long long)LSUM * C) return;
  int zo = (int)(e % C);
  long long lmn = e / C;
  int l = decode_l((int)lmn);
  int rem = (int)lmn - l_off(l), dim = 2 * l + 1;
  int ni = rem % dim;
  int z = zo / Fo, o = zo % Fo;
  int mu = (ni - l) & 63;
  float sr = 0.f, si = 0.f;
  for (int f = 0; f < Fin; f++) {
    long long gi = (lmn * (BATCH * Fin) + (long long)z * Fin + f) * 2;
    float ar = g[gi], ai = g[gi + 1];
    int kidx = (f * Fo + o) * 64 + mu;
    float cr = khre[kidx], ci = khim[kidx];
    sr += ar * cr + ai * ci;
    si += ai * cr - ar * ci;
  }
  zs[2 * e] = sr;
  zs[2 * e + 1] = si;
}

// T[z,o,j,u,v] = sum_l (2l+1)*dq[l][j,fu+l,fv+l] * zs[l,...]  (freq scatter)
__global__ void k_synthT(const float* zs, const float* dq, float* tre,
                         float* tim, int Fo) {
  long long idx = (long long)blockIdx.x * blockDim.x + threadIdx.x;
  long long tot = (long long)BATCH * Fo * VOL;
  if (idx >= tot) return;
  int v = (int)(idx & 63);
  int u = (int)((idx >> 6) & 63);
  int j = (int)((idx >> 12) & 63);
  int zo = (int)(idx >> 18);
  int o = zo % Fo, z = zo / Fo;
  int fu = (u < 32) ? u : u - 64;
  int fv = (v < 32) ? v : v - 64;
  int au = fu < 0 ? -fu : fu, av = fv < 0 ? -fv : fv;
  int lmin = au > av ? au : av;
  float ar = 0.f, ai = 0.f;
  for (int l = lmin; l < 32; l++) {
    int dim = 2 * l + 1;
    int base = l_off(l) + (fu + l) * dim + (fv + l);
    float ds = (float)dim * dq[(size_t)j * LSUM + base];
    long long zi = ((long long)base * (BATCH * Fo) + (long long)z * Fo + o) * 2;
    ar += ds * zs[zi];
    ai += ds * zs[zi + 1];
  }
  tre[idx] = ar;
  tim[idx] = ai;
}

// --------------------- WMMA batched 64x64 2-D transforms --------------------
// One plane per workgroup, 256 threads = 8 wave32. Stage 1: T1 = X * W^T,
// Stage 2: Y = W * T1. A-frag: lane<16 holds M=lane, K pair k0..k0+1; lane>=16
// holds K pair k0+2..k0+3. B-frag: VGPR0 = K (0|2), VGPR1 = K (1|3), N=lane&15.
// C/D: VGPR r = row r (lanes<16) / r+8 (lanes>=16), col = lane&15.
// wnim = element-wise negation of wim (the other DFT table), avoiding v_xor.

__global__ __launch_bounds__(256) void k_ifft2_real(
    const float* tre, const float* tim, const float* wre, const float* wim,
    const float* wnim, const float* bias, int Fo, float* act) {
  __shared__ __align__(16) float Xr[4096], Xi[4096], Sr[4096], Si[4096];
  int p = blockIdx.x;
  int o = (p / 64) % Fo;
  const float* gr = tre + (size_t)p * 4096;
  const float* gi = tim + (size_t)p * 4096;
#if HAS_ASYNC_LDS
  for (int i = threadIdx.x * 4; i < 4096; i += 1024) {
    async_cp16(gr + i, Xr + i);
    async_cp16(gi + i, Xi + i);
  }
  __builtin_amdgcn_s_wait_asynccnt(0);
  __syncthreads();
#else
  for (int i = threadIdx.x; i < 4096; i += 256) {
    Xr[i] = gr[i];
    Xi[i] = gi[i];
  }
  __syncthreads();
#endif
  int lane = threadIdx.x & 31, w = threadIdx.x >> 5;
  int tj = w >> 1, h = w & 1;
  // stage 1: S[m,g] = sum_n X[m,n] * W[g,n]   (complex)
  for (int s = 0; s < 2; s++) {
    int ti = h * 2 + s;
    v8f cr = {}, ci = {};
    int row = ti * 16 + (lane & 15);
    int col = tj * 16 + (lane & 15);
    for (int k0 = 0; k0 < 64; k0 += 4) {
      int kb = k0 + ((lane >> 4) << 1);
      v2f ar, am, br, bm, na;
      ar.x = Xr[row * 64 + kb];   ar.y = Xr[row * 64 + kb + 1];
      am.x = Xi[row * 64 + kb];   am.y = Xi[row * 64 + kb + 1];
      br.x = wre[col * 64 + kb];  br.y = wre[col * 64 + kb + 1];
      bm.x = wim[col * 64 + kb];  bm.y = wim[col * 64 + kb + 1];
      na.x = -am.x; na.y = -am.y;
      cr = wmma4(ar, br, cr); cr = wmma4(na, bm, cr);
      ci = wmma4(ar, bm, ci); ci = wmma4(am, br, ci);
    }
    for (int r = 0; r < 8; r++) {
      int rr = ti * 16 + r + ((lane >> 4) << 3);
      int cc = tj * 16 + (lane & 15);
      Sr[rr * 64 + cc] = cr[r];
      Si[rr * 64 + cc] = ci[r];
    }
  }
  __syncthreads();
  // stage 2: Y_re[a,g] = sum_m Re(W[a,m] * S[m,g]),  + bias
  float bo = bias ? bias[o] : 0.f;
  float* out = act + (size_t)p * 4096;
  for (int s = 0; s < 2; s++) {
    int t = w * 2 + s;
    int ti = t >> 2, tjj = t & 3;
    v8f cr = {};
    int row = ti * 16 + (lane & 15);
    int col = tjj * 16 + (lane & 15);
    for (int k0 = 0; k0 < 64; k0 += 4) {
      int kb = k0 + ((lane >> 4) << 1);
      v2f ar, na, br, bm;
      ar.x = wre[row * 64 + kb];   ar.y = wre[row * 64 + kb + 1];
      na.x = wnim[row * 64 + kb];  na.y = wnim[row * 64 + kb + 1];  // = -Im(W)
      br.x = Sr[kb * 64 + col];    br.y = Sr[(kb + 1) * 64 + col];
      bm.x = Si[kb * 64 + col];    bm.y = Si[(kb + 1) * 64 + col];
      cr = wmma4(ar, br, cr); cr = wmma4(na, bm, cr);
    }
    for (int r = 0; r < 8; r++) {
      int rr = ti * 16 + r + ((lane >> 4) << 3);
      int cc = tjj * 16 + (lane & 15);
      out[rr * 64 + cc] = cr[r] + bo;
    }
  }
}

__global__ __launch_bounds__(256) void k_fft2_fwd(
    const float* act, const float* wre, const float* wim, const float* wnim,
    float* xre, float* xim) {
  __shared__ __align__(16) float Xr[4096], Sr[4096], Si[4096];
  int p = blockIdx.x;
  const float* g = act + (size_t)p * 4096;
#if HAS_ASYNC_LDS
  for (int i = threadIdx.x * 4; i < 4096; i += 1024) async_cp16(g + i, Xr + i);
  __builtin_amdgcn_s_wait_asynccnt(0);
  __syncthreads();
#else
  for (int i = threadIdx.x; i < 4096; i += 256) Xr[i] = g[i];
  __syncthreads();
#endif
  int lane = threadIdx.x & 31, w = threadIdx.x >> 5;
  int tj = w >> 1, h = w & 1;
  // stage 1: S[a,v] = sum_g X[a,g] * W[v,g]   (real * complex)
  for (int s = 0; s < 2; s++) {
    int ti = h * 2 + s;
    v8f cr = {}, ci = {};
    int row = ti * 16 + (lane & 15);
    int col = tj * 16 + (lane & 15);
    for (int k0 = 0; k0 < 64; k0 += 4) {
      int kb = k0 + ((lane >> 4) << 1);
      v2f a, br, bm;
      a.x = Xr[row * 64 + kb];    a.y = Xr[row * 64 + kb + 1];
      br.x = wre[col * 64 + kb];  br.y = wre[col * 64 + kb + 1];
      bm.x = wim[col * 64 + kb];  bm.y = wim[col * 64 + kb + 1];
      cr = wmma4(a, br, cr);
      ci = wmma4(a, bm, ci);
    }
    for (int r = 0; r < 8; r++) {
      int rr = ti * 16 + r + ((lane >> 4) << 3);
      int cc = tj * 16 + (lane & 15);
      Sr[rr * 64 + cc] = cr[r];
      Si[rr * 64 + cc] = ci[r];
    }
  }
  __syncthreads();
  // stage 2: xf[u,v] = sum_a W[u,a] * S[a,v]   (complex * complex)
  float* orr = xre + (size_t)p * 4096;
  float* oi = xim + (size_t)p * 4096;
  for (int s = 0; s < 2; s++) {
    int t = w * 2 + s;
    int ti = t >> 2, tjj = t & 3;
    v8f cr = {}, ci = {};
    int row = ti * 16 + (lane & 15);
    int col = tjj * 16 + (lane & 15);
    for (int k0 = 0; k0 < 64; k0 += 4) {
      int kb = k0 + ((lane >> 4) << 1);
      v2f ar, am, na, br, bm;
      ar.x = wre[row * 64 + kb];   ar.y = wre[row * 64 + kb + 1];
      am.x = wim[row * 64 + kb];   am.y = wim[row * 64 + kb + 1];
      na.x = wnim[row * 64 + kb];  na.y = wnim[row * 64 + kb + 1];  // = -Im(W)
      br.x = Sr[kb * 64 + col];    br.y = Sr[(kb + 1) * 64 + col];
      bm.x = Si[kb * 64 + col];    bm.y = Si[(kb + 1) * 64 + col];
      cr = wmma4(ar, br, cr); cr = wmma4(na, bm, cr);
      ci = wmma4(ar, bm, ci); ci = wmma4(am, br, ci);
    }
    for (int r = 0; r < 8; r++) {
      int rr = ti * 16 + r + ((lane >> 4) << 3);
      int cc = tjj * 16 + (lane & 15);
      orr[rr * 64 + cc] = cr[r];
      oi[rr * 64 + cc] = ci[r];
    }
  }
}

// ------------------------------ BN/ReLU/mean -------------------------------

__global__ void k_bnstats(const float* act, int F, float* stats) {
  __shared__ double ssum[256], ssq[256];
  int o = blockIdx.x;
  double s = 0.0, s2 = 0.0;
  for (int z = 0; z < BATCH; z++) {
    const float* p = act + (size_t)(z * F + o) * VOL;
    for (int i = threadIdx.x; i < VOL; i += 256) {
      double v = p[i];
      s += v;
      s2 += v * v;
    }
  }
  ssum[threadIdx.x] = s;
  ssq[threadIdx.x] = s2;
  __syncthreads();
  for (int st = 128; st > 0; st >>= 1) {
    if (threadIdx.x < st) {
      ssum[threadIdx.x] += ssum[threadIdx.x + st];
      ssq[threadIdx.x] += ssq[threadIdx.x + st];
    }
    __syncthreads();
  }
  if (threadIdx.x == 0) {
    double n = (double)BATCH * VOL;
    double mu = ssum[0] / n;
    double var = ssq[0] / n - mu * mu;
    stats[2 * o] = (float)mu;
    stats[2 * o + 1] = (float)var;
  }
}

__global__ void k_bnrelu(float* act, const float* stats, const float* gam,
                         const float* bet, int F) {
  long long idx = (long long)blockIdx.x * blockDim.x + threadIdx.x;
  if (idx >= (long long)BATCH * F * VOL) return;
  int o = (int)((idx / VOL) % F);
  float mu = stats[2 * o], var = stats[2 * o + 1];
  float y = (act[idx] - mu) * rsqrtf(var + 1e-5f);
  y = y * gam[o] + bet[o];
  act[idx] = y > 0.f ? y : 0.f;
}

// one wave32 per output element: coalesced 256B reads + shuffle reduction
__global__ void k_mean_out(const float* act, float* out) {
  int lane = threadIdx.x & 31;
  int widx = (int)((blockIdx.x * (unsigned)blockDim.x + threadIdx.x) >> 5);
  if (widx >= BATCH * F2c * 64 * 64) return;
  const float* p = act + (size_t)widx * 64;
  float s = p[lane] + p[lane + 32];
  for (int m = 16; m > 0; m >>= 1) s += __shfl_xor(s, m, 32);
  if (lane == 0) out[widx] = s * (1.0f / 64.0f);
}

// ------------------------------- launcher ----------------------------------

static inline unsigned nblk(long long n) { return (unsigned)((n + 255) / 256); }

extern "C" void kernel_launch(void* const* d_in, const int* in_sizes, int n_in,
                              void* d_out, int out_size, void* d_ws,
                              size_t ws_size, hipStream_t stream) {
  const float* x = (const float*)d_in[0];
  const float* k1 = (const float*)d_in[1];
  const float* bias1 = (const float*)d_in[2];
  const float* g1 = (const float*)d_in[3];
  const float* be1 = (const float*)d_in[4];
  const float* k2 = (const float*)d_in[5];
  const float* bias2 = (const float*)d_in[6];
  const float* g2 = (const float*)d_in[7];
  const float* be2 = (const float*)d_in[8];
  const float* k3 = (const float*)d_in[9];
  const float* bias3 = (const float*)d_in[10];
  const float* g3 = (const float*)d_in[11];
  const float* be3 = (const float*)d_in[12];

  float* W = (float*)d_ws;
  const size_t LS = LSUM;
  float* dq = W;                         // 64*LS
  float* de = dq + 64 * LS;              // LS
  float* wgt = de + LS;                  // 64
  float* wfre = wgt + 64;                // 4096
  float* wfim = wfre + 4096;
  float* wire = wfim + 4096;
  float* wiim = wire + 4096;
  float* khre = wiim + 4096;             // up to 60*60*64
  float* khim = khre + 230400;
  float* x1re = khim + 230400;           // 24576
  float* x1im = x1re + 24576;
  float* fh1 = x1im + 24576;             // 12288
  float* fhb = fh1 + 12288;              // LS*120*2
  float* gb = fhb + LS * 120 * 2;        // LS*120*2
  float* zsb = gb + LS * 120 * 2;        // LS*120*2
  float* Tre = zsb + LS * 120 * 2;       // 120*VOL
  float* Tim = Tre + (size_t)120 * VOL;
  float* actb = Tim + (size_t)120 * VOL;
  float* xfre = actb + (size_t)120 * VOL;
  float* xfim = xfre + (size_t)120 * VOL;
  float* stats = xfim + (size_t)120 * VOL;  // 120

  const float s2s = 1.0f / sqrtf(196608.0f);  // 1/sqrt(N*F0*BW^2)
  const float s3s = 1.0f / sqrtf(3840.0f);    // 1/sqrt(N*F1)

  // constants
  k_wigner<<<65, 256, 0, stream>>>(dq, de);
  k_dhw<<<1, 64, 0, stream>>>(wgt);
  k_twiddle<<<16, 256, 0, stream>>>(wfre, wfim, wire, wiim);

  // ---- layer 1: S2 conv ----
  k_dft_alpha<<<nblk(24576), 256, 0, stream>>>(x, wfre, wfim, x1re, x1im);
  k_khat<<<nblk(3 * 60 * 64), 256, 0, stream>>>(k1, 3, 60, s2s, wfre, wfim,
                                                khre, khim);
  k_fh_s2<<<nblk(1024 * 6), 256, 0, stream>>>(x1re, x1im, dq, wgt, fh1);
  k_zs_s2<<<nblk((long long)LS * 120), 256, 0, stream>>>(fh1, de, khre, khim,
                                                         zsb);
  k_synthT<<<nblk((long long)120 * VOL), 256, 0, stream>>>(zsb, dq, Tre, Tim,
                                                           60);
  k_ifft2_real<<<7680, 256, 0, stream>>>(Tre, Tim, wire, wiim, wfim, bias1, 60,
                                         actb);
  k_bnstats<<<60, 256, 0, stream>>>(actb, 60, stats);
  k_bnrelu<<<nblk((long long)120 * VOL), 256, 0, stream>>>(actb, stats, g1, be1,
                                                           60);

  // ---- layer 2: SO3 conv ----
  k_fft2_fwd<<<7680, 256, 0, stream>>>(actb, wfre, wfim, wiim, xfre, xfim);
  k_khat<<<nblk(60 * 60 * 64), 256, 0, stream>>>(k2, 60, 60, s3s, wfre, wfim,
                                                 khre, khim);
  k_fh_so3<<<nblk((long long)LS * 120), 256, 0, stream>>>(xfre, xfim, dq, wgt,
                                                          60, fhb);
  k_g_so3<<<nblk((long long)LS * 120), 256, 0, stream>>>(fhb, de, 60, gb);
  k_zs_so3<<<nblk((long long)LS * 120), 256, 0, stream>>>(gb, khre, khim, 60,
                                                          60, zsb);
  k_synthT<<<nblk((long long)120 * VOL), 256, 0, stream>>>(zsb, dq, Tre, Tim,
                                                           60);
  k_ifft2_real<<<7680, 256, 0, stream>>>(Tre, Tim, wire, wiim, wfim, bias2, 60,
                                         actb);
  k_bnstats<<<60, 256, 0, stream>>>(actb, 60, stats);
  k_bnrelu<<<nblk((long long)120 * VOL), 256, 0, stream>>>(actb, stats, g2, be2,
                                                           60);

  // ---- layer 3: SO3 conv ----
  k_fft2_fwd<<<7680, 256, 0, stream>>>(actb, wfre, wfim, wiim, xfre, xfim);
  k_khat<<<nblk(60 * 36 * 64), 256, 0, stream>>>(k3, 60, 36, s3s, wfre, wfim,
                                                 khre, khim);
  k_fh_so3<<<nblk((long long)LS * 120), 256, 0, stream>>>(xfre, xfim, dq, wgt,
                                                          60, fhb);
  k_g_so3<<<nblk((long long)LS * 120), 256, 0, stream>>>(fhb, de, 60, gb);
  k_zs_so3<<<nblk((long long)LS * 72), 256, 0, stream>>>(gb, khre, khim, 60, 36,
                                                         zsb);
  k_synthT<<<nblk((long long)72 * VOL), 256, 0, stream>>>(zsb, dq, Tre, Tim,
                                                          36);
  k_ifft2_real<<<4608, 256, 0, stream>>>(Tre, Tim, wire, wiim, wfim, bias3, 36,
                                         actb);
  k_bnstats<<<36, 256, 0, stream>>>(actb, 36, stats);
  k_bnrelu<<<nblk((long long)72 * VOL), 256, 0, stream>>>(actb, stats, g3, be3,
                                                          36);

  k_mean_out<<<nblk(294912LL * 32), 256, 0, stream>>>(actb, (float*)d_out);
}